// Node_12695923327542
// MI455X (gfx1250) — compile-verified
//
#include <hip/hip_runtime.h>
#include <hip/hip_bf16.h>

// Soft decision tree forward:  out[b] = sum_leaf prod_path(w / 1-w) * leaf_value
// Memory-bound (~790 MB -> ~34 us at 23.3 TB/s). One wave32 per batch row.

#define TREE_DEPTH 8
constexpr int N_INT  = 255;   // internal nodes
constexpr int N_LEAF = 256;
constexpr int N_FEAT = 256;
constexpr int ROWS_PER_BLOCK = 8;   // 8 waves * 32 lanes = 256 threads

#if defined(__HIP_DEVICE_COMPILE__) && __has_builtin(__builtin_amdgcn_global_load_async_to_lds_b128)
#define USE_ASYNC_BUILTIN 1
#else
#define USE_ASYNC_BUILTIN 0
#endif

typedef __attribute__((__vector_size__(16))) int async_i4;

__device__ __forceinline__ void async_x_to_lds(const float* gsrc, float* lds_dst) {
#if USE_ASYNC_BUILTIN
  __builtin_amdgcn_global_load_async_to_lds_b128(
      (__attribute__((address_space(1))) async_i4*)gsrc,
      (__attribute__((address_space(3))) async_i4*)lds_dst, 0, 0);
#elif defined(__HIP_DEVICE_COMPILE__)
  // Fallback: raw CDNA5 async-load encoding. LDS generic pointers carry the
  // LDS byte offset in their low 32 bits (ISA 10.2 aperture mapping).
  unsigned int lds_off = (unsigned int)(size_t)lds_dst;
  asm volatile("global_load_async_to_lds_b128 %0, %1, off"
               :: "v"(lds_off), "v"(gsrc) : "memory");
#else
  (void)gsrc; (void)lds_dst;
#endif
}

__global__ __launch_bounds__(256) void soft_tree_kernel(
    const float* __restrict__ x,           // (B, 256)
    const float* __restrict__ split_cond,  // (255,)
    const float* __restrict__ u,           // (B, 255, 2)
    const float* __restrict__ leaf_values, // (256,)
    const int*   __restrict__ split_idx,   // (255,)
    float*       __restrict__ out,         // (B,)
    int batch)
{
  __shared__ float s_x[ROWS_PER_BLOCK][N_FEAT];   // 8 KB  staged x rows
  __shared__ float s_w[ROWS_PER_BLOCK][N_LEAF];   // 8 KB  gate probabilities
  __shared__ float s_sc[N_LEAF];                  // 1 KB  split_cond
  __shared__ int   s_si[N_LEAF];                  // 1 KB  split_idx

  const int t = threadIdx.x;
  if (t < N_INT) { s_sc[t] = split_cond[t]; s_si[t] = split_idx[t]; }
  __syncthreads();

  const int wave = t >> 5;
  const int lane = t & 31;
  const int row  = blockIdx.x * ROWS_PER_BLOCK + wave;
  if (row >= batch) return;

  // ---- stage x row (1 KB) into LDS: 2 x async b128 per lane ----
  const float* __restrict__ xrow = x + (size_t)row * N_FEAT;
  async_x_to_lds(xrow + lane * 4,        &s_x[wave][lane * 4]);
  async_x_to_lds(xrow + lane * 4 + 128,  &s_x[wave][lane * 4 + 128]);
  asm volatile("s_wait_asynccnt 0" ::: "memory");

  // ---- compute 255 gates; u streamed coalesced (float2 per node) ----
  const float2* __restrict__ u2 = (const float2*)(u + (size_t)row * (2 * N_INT));
#pragma unroll
  for (int it = 0; it < 8; ++it) {
    const int n = lane + 32 * it;
    if (n < N_INT) {
      float2 uv = u2[n];
      float u0 = fminf(fmaxf(uv.x, 1e-10f), 1.0f);
      float u1 = fminf(fmaxf(uv.y, 1e-10f), 1.0f);
      float g0 = -__logf(-__logf(u0));          // gumbel
      float g1 = -__logf(-__logf(u1));
      float c  = s_x[wave][s_si[n]] - s_sc[n] + (g0 - g1);
      float w  = __builtin_amdgcn_rcpf(1.0f + __expf(-c));   // sigmoid
      s_w[wave][n] = w;
    }
  }
  asm volatile("s_wait_dscnt 0" ::: "memory");  // wave-lockstep: gates visible

  // ---- bottom-up collapse: V(node) = Vl + w*(Vr - Vl) ----
  // lane holds leaves [8*lane, 8*lane+8)
  const float4* __restrict__ lf4 = (const float4*)leaf_values;
  float4 L0 = lf4[lane * 2];
  float4 L1 = lf4[lane * 2 + 1];
  float L[8] = {L0.x, L0.y, L0.z, L0.w, L1.x, L1.y, L1.z, L1.w};

  const float* __restrict__ wr = &s_w[wave][0];

  float v7[4];                                  // level 7: nodes 127..254
#pragma unroll
  for (int j = 0; j < 4; ++j) {
    float w = wr[127 + 4 * lane + j];
    v7[j] = fmaf(w, L[2 * j + 1] - L[2 * j], L[2 * j]);
  }
  float v6[2];                                  // level 6: nodes 63..126
#pragma unroll
  for (int j = 0; j < 2; ++j) {
    float w = wr[63 + 2 * lane + j];
    v6[j] = fmaf(w, v7[2 * j + 1] - v7[2 * j], v7[2 * j]);
  }
  float v = fmaf(wr[31 + lane], v6[1] - v6[0], v6[0]);  // level 5: node=lane

  // levels 4..0 via wave32 butterfly shuffles
  const int offs[5] = {15, 7, 3, 1, 0};
#pragma unroll
  for (int k = 0; k < 5; ++k) {
    const int stride = 1 << k;
    float other = __shfl_xor(v, stride, 32);
    float vl = (lane & stride) ? other : v;
    float vr = (lane & stride) ? v : other;
    float w  = wr[offs[k] + (lane >> (k + 1))];
    v = fmaf(w, vr - vl, vl);
  }

  if (lane == 0) out[row] = v;
}

extern "C" void kernel_launch(void* const* d_in, const int* in_sizes, int n_in,
                              void* d_out, int out_size, void* d_ws, size_t ws_size,
                              hipStream_t stream) {
  (void)n_in; (void)out_size; (void)d_ws; (void)ws_size;
  const float* x    = (const float*)d_in[0];
  const float* sc   = (const float*)d_in[1];
  const float* u    = (const float*)d_in[2];
  const float* leaf = (const float*)d_in[3];
  const int*   si   = (const int*)d_in[4];
  float* out        = (float*)d_out;

  const int batch = in_sizes[0] / N_FEAT;
  dim3 grid((batch + ROWS_PER_BLOCK - 1) / ROWS_PER_BLOCK);
  dim3 block(256);
  hipLaunchKernelGGL(soft_tree_kernel, grid, block, 0, stream,
                     x, sc, u, leaf, si, out, batch);
}